// GCN_Net0_43052752175664
// MI455X (gfx1250) — compile-verified
//
#include <hip/hip_runtime.h>
#include <hip/hip_bf16.h>

#define N_IN_F  128
#define N_OUT_F 32

typedef __attribute__((ext_vector_type(2))) float v2f;
typedef __attribute__((ext_vector_type(8))) float v8f;

// ---------------------------------------------------------------------------
// 1) deg init: deg[i] = 1.0 (self-loop)
// ---------------------------------------------------------------------------
__global__ void deg_init_kernel(float* __restrict__ deg, int n) {
    int i = blockIdx.x * blockDim.x + threadIdx.x;
    if (i < n) deg[i] = 1.0f;
}

// ---------------------------------------------------------------------------
// 2) deg accumulate: deg[dst] += 1 for every edge (dst = edge_index[1][e])
// ---------------------------------------------------------------------------
__global__ void deg_acc_kernel(const long long* __restrict__ ei, long long E,
                               float* __restrict__ deg) {
    long long e = (long long)blockIdx.x * blockDim.x + threadIdx.x;
    if (e < E) {
        long long dst = ei[E + e];
        atomicAdd(&deg[dst], 1.0f);
    }
}

// ---------------------------------------------------------------------------
// 3) dis[i] = rsqrt(deg[i])   (deg >= 1 always, so no zero guard needed)
// ---------------------------------------------------------------------------
__global__ void deg_rsqrt_kernel(float* __restrict__ deg, int n) {
    int i = blockIdx.x * blockDim.x + threadIdx.x;
    if (i < n) deg[i] = rsqrtf(deg[i]);
}

// ---------------------------------------------------------------------------
// 4) h = x @ W^T  via V_WMMA_F32_16X16X4_F32
//    Block: 256 threads = 8 waves.  Block covers 64 node rows.
//    Wave w: row tile (w>>1)*16, col tile (w&1)*16.  K = 128 in 32 chunks of 4.
// ---------------------------------------------------------------------------
__launch_bounds__(256)
__global__ void gcn_gemm_wmma_kernel(const float* __restrict__ x,
                                     const float* __restrict__ W,
                                     float* __restrict__ h, int nNodes) {
    // sW[k*32 + o] = W[o*128 + k]   (B-matrix, K-major)
    __shared__ float sW[N_IN_F * N_OUT_F];          // 16 KB
    // x tile, padded row stride 132 floats to kill bank conflicts
    __shared__ float sX[64 * 132];                  // ~33 KB

    const int tid = threadIdx.x;
    const int rowBase = blockIdx.x * 64;

    // ---- stage W transposed into LDS (coalesced global reads) ----
    #pragma unroll
    for (int t = 0; t < 16; ++t) {
        int idx = tid + t * 256;        // 0..4095
        int o = idx >> 7;               // / 128
        int k = idx & 127;
        sW[k * N_OUT_F + o] = W[idx];
    }

    // ---- stage 64x128 x tile into LDS via float4 (coalesced) ----
    #pragma unroll
    for (int t = 0; t < 8; ++t) {
        int idx = tid + t * 256;        // float4 index, 0..2047
        int row = idx >> 5;             // 32 float4 per row
        int c4  = idx & 31;
        float4 v = make_float4(0.f, 0.f, 0.f, 0.f);
        int grow = rowBase + row;
        if (grow < nNodes) v = ((const float4*)x)[(size_t)grow * 32 + c4];
        *(float4*)&sX[row * 132 + c4 * 4] = v;      // 132*4B row stride, 16B aligned
    }
    __syncthreads();

    const int lane = tid & 31;
    const int wid  = tid >> 5;
    const int rowLocal = (wid >> 1) * 16;   // 0,16,32,48
    const int colBase  = (wid & 1)  * 16;   // 0 or 16
    const int half = lane >> 4;             // 0: K=0,1 ; 1: K=2,3
    const int l15  = lane & 15;

    v8f acc = {0.f, 0.f, 0.f, 0.f, 0.f, 0.f, 0.f, 0.f};
    const float* aRow = &sX[(rowLocal + l15) * 132 + 2 * half];

    #pragma unroll
    for (int c = 0; c < 32; ++c) {
        const int k0 = c * 4;
        v2f a, b;
        // A 16x4 layout: lanes 0-15 -> K=k0,k0+1 ; lanes 16-31 -> K=k0+2,k0+3
        a.x = aRow[k0];
        a.y = aRow[k0 + 1];
        // B 4x16 layout mirrors A: lanes 0-15 -> K=k0,k0+1 at N=lane
        const int kb = k0 + 2 * half;
        b.x = sW[kb * N_OUT_F + colBase + l15];
        b.y = sW[(kb + 1) * N_OUT_F + colBase + l15];
        acc = __builtin_amdgcn_wmma_f32_16x16x4_f32(
            /*neg_a=*/false, a, /*neg_b=*/false, b,
            /*c_mod=*/(short)0, acc, /*reuse_a=*/false, /*reuse_b=*/false);
    }

    // D 16x16 layout: lanes 0-15 hold M=r, lanes 16-31 hold M=r+8, N=lane&15
    #pragma unroll
    for (int r = 0; r < 8; ++r) {
        int grow = rowBase + rowLocal + r + 8 * half;
        if (grow < nNodes)
            h[(size_t)grow * N_OUT_F + colBase + l15] = acc[r];
    }
}

// ---------------------------------------------------------------------------
// 5) out[i][f] = b[f] + h[i][f] * dis[i]^2   (bias + self-loop message)
// ---------------------------------------------------------------------------
__global__ void out_init_kernel(const float* __restrict__ h,
                                const float* __restrict__ dis,
                                const float* __restrict__ b,
                                float* __restrict__ out, int n) {
    long long t = (long long)blockIdx.x * blockDim.x + threadIdx.x;
    if (t < (long long)n * N_OUT_F) {
        int i = (int)(t >> 5);      // / 32
        int f = (int)(t & 31);
        float d = dis[i];
        out[t] = b[f] + h[t] * d * d;
    }
}

// ---------------------------------------------------------------------------
// 6) edge scatter: out[dst] += h[src] * dis[src]*dis[dst]
//    One wave per edge: 32 lanes = 32 features -> coalesced row read + atomic row add.
// ---------------------------------------------------------------------------
__global__ void edge_scatter_kernel(const long long* __restrict__ ei, long long E,
                                    const float* __restrict__ h,
                                    const float* __restrict__ dis,
                                    float* __restrict__ out) {
    long long t = (long long)blockIdx.x * blockDim.x + threadIdx.x;
    long long e = t >> 5;
    int f = (int)(t & 31);
    if (e < E) {
        long long s = ei[e];
        long long d = ei[E + e];
        float norm = dis[s] * dis[d];
        float val = h[s * N_OUT_F + f] * norm;
        atomicAdd(&out[d * N_OUT_F + f], val);
    }
}

// ---------------------------------------------------------------------------
// launcher
// ---------------------------------------------------------------------------
extern "C" void kernel_launch(void* const* d_in, const int* in_sizes, int n_in,
                              void* d_out, int out_size, void* d_ws, size_t ws_size,
                              hipStream_t stream) {
    const float*     x  = (const float*)d_in[0];
    const long long* ei = (const long long*)d_in[1];   // int64 edge_index [2, E]
    const float*     W  = (const float*)d_in[2];
    const float*     b  = (const float*)d_in[3];
    float*           out = (float*)d_out;

    const int       N = in_sizes[0] / N_IN_F;          // 100000
    const long long E = (long long)in_sizes[1] / 2;    // 1,600,000

    // workspace: [dis: N floats][h: N*32 floats]
    char* ws = (char*)d_ws;
    float* dis = (float*)ws;
    size_t hOff = (((size_t)N * sizeof(float)) + 255) & ~(size_t)255;
    float* h = (float*)(ws + hOff);

    const int T = 256;

    deg_init_kernel<<<(N + T - 1) / T, T, 0, stream>>>(dis, N);
    deg_acc_kernel<<<(unsigned)((E + T - 1) / T), T, 0, stream>>>(ei, E, dis);
    deg_rsqrt_kernel<<<(N + T - 1) / T, T, 0, stream>>>(dis, N);
    gcn_gemm_wmma_kernel<<<(N + 63) / 64, T, 0, stream>>>(x, W, h, N);

    long long nOut = (long long)N * N_OUT_F;
    out_init_kernel<<<(unsigned)((nOut + T - 1) / T), T, 0, stream>>>(h, dis, b, out, N);

    long long nEdgeWork = E * N_OUT_F;
    edge_scatter_kernel<<<(unsigned)((nEdgeWork + T - 1) / T), T, 0, stream>>>(ei, E, h, dis, out);
}